// GraphConvolution_1013612282172
// MI455X (gfx1250) — compile-verified
//
#include <hip/hip_runtime.h>
#include <hip/hip_bf16.h>

#define IN_DIM  512
#define OUT_DIM 256
#define LDA     516   // padded LDS row stride in floats (516 % 64 == 4 -> conflict-free)

typedef __attribute__((ext_vector_type(2))) float v2f;
typedef __attribute__((ext_vector_type(8))) float v8f;

// ---------------------------------------------------------------------------
// Dual GEMM: preA = x @ W0, preB = x @ W1, using V_WMMA_F32_16X16X4_F32.
// Block = 256 threads (8 waves). Each block owns 16 rows of x (staged in LDS)
// and produces all 512 output columns (256 for W0 -> preA, 256 for W1 -> preB).
// Wave w handles 16x16 output tiles t = w, w+8, w+16, w+24.
// ---------------------------------------------------------------------------
__global__ __launch_bounds__(256) void gemm_dual_wmma(
    const float* __restrict__ x,
    const float* __restrict__ W0,
    const float* __restrict__ W1,
    float* __restrict__ preA,
    float* __restrict__ preB)
{
    __shared__ float As[16 * LDA];
    const int row0 = blockIdx.x * 16;

    // Cooperative load of the 16 x 512 A tile (float4 per thread, 8 iters).
    for (int i = threadIdx.x; i < 16 * (IN_DIM / 4); i += 256) {
        const int r  = i / (IN_DIM / 4);
        const int c4 = i % (IN_DIM / 4);
        const float4 v = ((const float4*)(x + (size_t)(row0 + r) * IN_DIM))[c4];
        float* dst = As + r * LDA + c4 * 4;
        dst[0] = v.x; dst[1] = v.y; dst[2] = v.z; dst[3] = v.w;
    }
    __syncthreads();

    const int lane = threadIdx.x & 31;
    const int wave = threadIdx.x >> 5;
    const int m    = lane & 15;        // A: lane -> M row; B/C/D: lane -> N col
    const int half = lane >> 4;        // 0: K/M group 0, 1: K/M group 1
    const int koff = half << 1;        // A/B K offset within 4-wide step

    const float* arow = As + m * LDA + koff;

    for (int t = wave; t < 32; t += 8) {
        const float* W   = (t < 16) ? W0 : W1;
        float*       dst = (t < 16) ? preA : preB;
        const int    n0  = (t & 15) * 16;
        const float* bcol = W + n0 + m;      // column n = lane&15

        v8f acc = {};
        #pragma unroll 4
        for (int k = 0; k < IN_DIM; k += 4) {
            // A fragment: 16x4 f32, lane m, VGPR0 = K=koff, VGPR1 = K=koff+1
            v2f a = *(const v2f*)(arow + k);
            // B fragment: 4x16 f32, mirrored layout
            v2f b;
            b.x = bcol[(size_t)(k + koff)     * OUT_DIM];
            b.y = bcol[(size_t)(k + koff + 1) * OUT_DIM];
            acc = __builtin_amdgcn_wmma_f32_16x16x4_f32(
                /*neg_a=*/false, a, /*neg_b=*/false, b,
                /*c_mod=*/(short)0, acc, /*reuse_a=*/false, /*reuse_b=*/false);
        }

        // C/D layout: VGPR r -> M = r + 8*half, lane -> N = n0 + m
        float* o = dst + (size_t)row0 * OUT_DIM + n0 + m;
        #pragma unroll
        for (int r = 0; r < 8; ++r)
            o[(size_t)(r + 8 * half) * OUT_DIM] = acc[r];
    }
}

// ---------------------------------------------------------------------------
// SpMM scatter: one wave32 per edge. Lane l gathers pre[col][8l..8l+7]
// (two coalesced float4 loads -> 1KB per wave) and does 8 f32 atomic adds
// into out[row][...]. Handles both edge lists in one launch.
// ---------------------------------------------------------------------------
__global__ __launch_bounds__(256) void spmm_edges(
    const float* __restrict__ preA, const float* __restrict__ preB,
    const int* __restrict__ rows0, const int* __restrict__ cols0,
    const float* __restrict__ vals0,
    const int* __restrict__ rows1, const int* __restrict__ cols1,
    const float* __restrict__ vals1,
    float* __restrict__ out, int E)
{
    const int wgid = blockIdx.x * 8 + (threadIdx.x >> 5);   // global wave id
    if (wgid >= 2 * E) return;
    const int lane = threadIdx.x & 31;

    const float* pre;
    int e, r, c;
    float v;
    if (wgid < E) {
        e = wgid;     pre = preA;
        r = rows0[e]; c = cols0[e]; v = vals0[e];
    } else {
        e = wgid - E; pre = preB;
        r = rows1[e]; c = cols1[e]; v = vals1[e];
    }

    const float4* src = (const float4*)(pre + (size_t)c * OUT_DIM) + lane * 2;
    float*        dst = out + (size_t)r * OUT_DIM + lane * 8;

    const float4 a = src[0];
    const float4 b = src[1];
    atomicAdd(dst + 0, v * a.x);
    atomicAdd(dst + 1, v * a.y);
    atomicAdd(dst + 2, v * a.z);
    atomicAdd(dst + 3, v * a.w);
    atomicAdd(dst + 4, v * b.x);
    atomicAdd(dst + 5, v * b.y);
    atomicAdd(dst + 6, v * b.z);
    atomicAdd(dst + 7, v * b.w);
}

// ---------------------------------------------------------------------------
// In-place ReLU, float4 vectorized.
// ---------------------------------------------------------------------------
__global__ __launch_bounds__(256) void relu4(float4* __restrict__ out, int n4)
{
    const int i = blockIdx.x * 256 + threadIdx.x;
    if (i >= n4) return;
    float4 v = out[i];
    v.x = fmaxf(v.x, 0.0f);
    v.y = fmaxf(v.y, 0.0f);
    v.z = fmaxf(v.z, 0.0f);
    v.w = fmaxf(v.w, 0.0f);
    out[i] = v;
}

extern "C" void kernel_launch(void* const* d_in, const int* in_sizes, int n_in,
                              void* d_out, int out_size, void* d_ws, size_t ws_size,
                              hipStream_t stream)
{
    const float* x     = (const float*)d_in[0];
    const float* W0    = (const float*)d_in[1];
    const float* W1    = (const float*)d_in[2];
    const float* vals0 = (const float*)d_in[3];
    const float* vals1 = (const float*)d_in[4];
    const int*   rows0 = (const int*)d_in[5];
    const int*   cols0 = (const int*)d_in[6];
    const int*   rows1 = (const int*)d_in[7];
    const int*   cols1 = (const int*)d_in[8];
    float*       out   = (float*)d_out;

    const int N = in_sizes[0] / IN_DIM;     // 100000 (divisible by 16)
    const int E = in_sizes[3];              // 3200000

    float* preA = (float*)d_ws;
    float* preB = preA + (size_t)N * OUT_DIM;

    // Zero the accumulation target (graph-capture safe).
    hipMemsetAsync(d_out, 0, (size_t)out_size * sizeof(float), stream);

    // Dense part: pre0/pre1 via WMMA.
    gemm_dual_wmma<<<N / 16, 256, 0, stream>>>(x, W0, W1, preA, preB);

    // Sparse scatter-add over both edge lists.
    const long long waves = 2LL * E;
    const int spmm_blocks = (int)((waves + 7) / 8);
    spmm_edges<<<spmm_blocks, 256, 0, stream>>>(preA, preB,
                                                rows0, cols0, vals0,
                                                rows1, cols1, vals1,
                                                out, E);

    // Final ReLU.
    const int n4 = out_size / 4;
    relu4<<<(n4 + 255) / 256, 256, 0, stream>>>((float4*)d_out, n4);
}